// ETNNLayer_60696477827105
// MI455X (gfx1250) — compile-verified
//
#include <hip/hip_runtime.h>
#include <hip/hip_bf16.h>

typedef __attribute__((ext_vector_type(2))) float v2f;
typedef __attribute__((ext_vector_type(8))) float v8f;

#define NN   100000
#define EE   400000
#define HH   128
#define DINK 261
#define KPAD 264
#define MPAD 132
#define BNEPS 1e-5f
#define POSC  0.1f

__device__ __forceinline__ v8f wmma_f32(v2f a, v2f b, v8f c) {
  // D = A(16x4 f32) * B(4x16 f32) + C(16x16 f32)
  return __builtin_amdgcn_wmma_f32_16x16x4_f32(false, a, false, b, (short)0, c, false, false);
}

// LDS byte offset (from wave LDS base) of a generic pointer into __shared__
__device__ __forceinline__ unsigned lds_off32(const void* p) {
  return (unsigned)(uintptr_t)(__attribute__((address_space(3))) const char*)p;
}

// async DMA: 32 lanes x 16B global -> LDS, tracked by ASYNCcnt
__device__ __forceinline__ void async_b128(unsigned lds, const void* g) {
  asm volatile("global_load_async_to_lds_b128 %0, %1, off"
               :: "v"(lds), "v"((unsigned long long)(uintptr_t)g)
               : "memory");
}
__device__ __forceinline__ void wait_async0() {
  asm volatile("s_wait_asynccnt 0x0" ::: "memory");
}

// ---------------- zero ----------------
__global__ void zero_kernel(float* __restrict__ p, size_t n) {
  size_t i = (size_t)blockIdx.x * blockDim.x + threadIdx.x;
  size_t stride = (size_t)gridDim.x * blockDim.x;
  for (; i < n; i += stride) p[i] = 0.0f;
}

// ---------------- edge-BN column statistics (gathered) ----------------
// state columns: [0,128) = x_send[idx0], [128,256) = x_rec[idx1], [256,261) = inv
__global__ void edge_stats_kernel(const float* __restrict__ xs, const float* __restrict__ xr,
                                  const int* __restrict__ idx, const float* __restrict__ inv,
                                  float* __restrict__ sums, float* __restrict__ sqs, int E) {
  int t = threadIdx.x;  // blockDim = 288 (9 waves); threads >= 261 idle
  float s = 0.f, q = 0.f;
  for (int e = blockIdx.x; e < E; e += gridDim.x) {
    float v = 0.f;
    if (t < 128)        v = xs[(size_t)idx[e] * HH + t];
    else if (t < 256)   v = xr[(size_t)idx[E + e] * HH + (t - 128)];
    else if (t < DINK)  v = inv[(size_t)e * 5 + (t - 256)];
    s += v; q += v * v;
  }
  if (t < DINK) { atomicAdd(&sums[t], s); atomicAdd(&sqs[t], q); }
}

// ---------------- node-BN column statistics (2 or 3 concat parts) ----------------
__global__ void node_stats_kernel(const float* __restrict__ p0, const float* __restrict__ p1,
                                  const float* __restrict__ p2,
                                  float* __restrict__ sums, float* __restrict__ sqs, int N) {
  int t = threadIdx.x;  // blockDim = 128*parts
  const float* p = (t < 128) ? p0 : ((t < 256) ? p1 : p2);
  int c = t & 127;
  float s = 0.f, q = 0.f;
  for (int n = blockIdx.x; n < N; n += gridDim.x) {
    float v = p[(size_t)n * HH + c];
    s += v; q += v * v;
  }
  atomicAdd(&sums[t], s); atomicAdd(&sqs[t], q);
}

// ---------------- fold BN into Linear: W'[c,h]=W[c,h]*scale_c ; b'=bW+sum shift_c*W[c,:]
__global__ void fold_kernel(const float* __restrict__ W, const float* __restrict__ bW,
                            const float* __restrict__ g, const float* __restrict__ bb,
                            const float* __restrict__ sums, const float* __restrict__ sqs,
                            float invM, float* __restrict__ Wp, float* __restrict__ bp,
                            int K, int KP) {
  int h = threadIdx.x;  // 128 threads, 1 block
  float acc = bW[h];
  for (int c = 0; c < K; ++c) {
    float mean = sums[c] * invM;
    float var  = sqs[c] * invM - mean * mean;
    float sc   = g[c] * rsqrtf(var + BNEPS);
    float sh   = bb[c] - mean * sc;
    float w    = W[(size_t)c * HH + h];
    Wp[(size_t)c * HH + h] = w * sc;
    acc += sh * w;
  }
  for (int c = K; c < KP; ++c) Wp[(size_t)c * HH + h] = 0.f;
  bp[h] = acc;
}

// ---------------- message kernel: 1 wave per 32-edge tile, fp32 WMMA ----------------
__global__ __launch_bounds__(32) void msg_kernel(
    const float* __restrict__ xs, const float* __restrict__ xr,
    const int* __restrict__ idx, const float* __restrict__ inv,
    const float* __restrict__ Wp, const float* __restrict__ bp,
    const float* __restrict__ iW, const float* __restrict__ ib,
    float* __restrict__ mes, int E) {
  __shared__ __align__(16) float S[32][KPAD];   // gathered state tile (K-padded)
  __shared__ __align__(16) float Mt[32][MPAD];  // SiLU output tile (padded vs bank conflicts)
  const int lane = threadIdx.x;
  const int e0 = blockIdx.x * 32;

  const int rs_l = idx[e0 + lane];       // this lane's edge: send row
  const int rr_l = idx[E + e0 + lane];   // this lane's edge: rec row

  // async DMA the gathered x rows: one b128 op moves a full 512B row per matrix
  const unsigned s_base = lds_off32(&S[0][0]) + (unsigned)lane * 16u;
  for (int r = 0; r < 32; ++r) {
    const int rs = __shfl(rs_l, r);
    const int rr = __shfl(rr_l, r);
    async_b128(s_base + (unsigned)r * (KPAD * 4u),        xs + (size_t)rs * HH + lane * 4);
    async_b128(s_base + (unsigned)r * (KPAD * 4u) + 512u, xr + (size_t)rr * HH + lane * 4);
  }
  {  // invariants + K padding: lane owns its edge row
    const float* ep = inv + (size_t)(e0 + lane) * 5;
#pragma unroll
    for (int j = 0; j < 5; ++j) S[lane][256 + j] = ep[j];
    S[lane][261] = 0.f; S[lane][262] = 0.f; S[lane][263] = 0.f;
  }
  wait_async0();
  __syncthreads();

  const int m  = lane & 15;  // A row (within half-tile) / D col
  const int hf = lane >> 4;  // K-half / M-half selector
  v8f acc0[8], acc1[8];
#pragma unroll
  for (int i = 0; i < 8; ++i) { acc0[i] = (v8f)0.f; acc1[i] = (v8f)0.f; }

  for (int k0 = 0; k0 < KPAD; k0 += 4) {
    const int ka = k0 + 2 * hf;
    v2f a0; a0.x = S[m][ka];      a0.y = S[m][ka + 1];
    v2f a1; a1.x = S[16 + m][ka]; a1.y = S[16 + m][ka + 1];
#pragma unroll
    for (int nt = 0; nt < 8; ++nt) {
      const int n = nt * 16 + m;
      v2f b; b.x = Wp[(size_t)ka * HH + n]; b.y = Wp[(size_t)(ka + 1) * HH + n];
      acc0[nt] = wmma_f32(a0, b, acc0[nt]);   // B reused by both M half-tiles
      acc1[nt] = wmma_f32(a1, b, acc1[nt]);
    }
  }

  // epilogue: bias + SiLU -> LDS
#pragma unroll
  for (int nt = 0; nt < 8; ++nt) {
    const int n = nt * 16 + m;
    const float bias = bp[n];
#pragma unroll
    for (int r = 0; r < 8; ++r) {
      float xa = acc0[nt][r] + bias;
      Mt[r + 8 * hf][n]      = xa / (1.f + __expf(-xa));  // silu
      float xb = acc1[nt][r] + bias;
      Mt[16 + r + 8 * hf][n] = xb / (1.f + __expf(-xb));
    }
  }
  __syncthreads();

  // per-edge sigmoid gate: each lane owns one edge (row stride MPAD avoids bank conflicts)
  float part = 0.f;
  for (int c = 0; c < HH; ++c) part += Mt[lane][c] * iW[c];
  const float g_l = 1.f / (1.f + __expf(-(part + ib[0])));

  // gated scatter-add into mes[idx1]
  for (int r = 0; r < 32; ++r) {
    const int rr = __shfl(rr_l, r);
    const float gr = __shfl(g_l, r);
    for (int c = lane; c < HH; c += 32)
      atomicAdd(&mes[(size_t)rr * HH + c], Mt[r][c] * gr);
  }
}

// ---------------- update kernel: residual + BN-folded GEMM, fp32 WMMA ----------------
__global__ __launch_bounds__(32) void update_kernel(
    const float* __restrict__ p0, const float* __restrict__ p1, const float* __restrict__ p2,
    const float* __restrict__ Wp, const float* __restrict__ bp,
    float* __restrict__ out, int K) {
  __shared__ __align__(16) float S[32][384];
  const int lane = threadIdx.x;
  const int n0 = blockIdx.x * 32;

  const unsigned s_base = lds_off32(&S[0][0]) + (unsigned)lane * 16u;
  for (int r = 0; r < 32; ++r) {
    const size_t n = (size_t)(n0 + r);
    async_b128(s_base + (unsigned)r * (384u * 4u),         p0 + n * HH + lane * 4);
    async_b128(s_base + (unsigned)r * (384u * 4u) + 512u,  p1 + n * HH + lane * 4);
    if (K > 256)
      async_b128(s_base + (unsigned)r * (384u * 4u) + 1024u, p2 + n * HH + lane * 4);
  }
  wait_async0();
  __syncthreads();

  const int m  = lane & 15;
  const int hf = lane >> 4;
  v8f acc0[8], acc1[8];
#pragma unroll
  for (int i = 0; i < 8; ++i) { acc0[i] = (v8f)0.f; acc1[i] = (v8f)0.f; }

  for (int k0 = 0; k0 < K; k0 += 4) {
    const int ka = k0 + 2 * hf;
    v2f a0; a0.x = S[m][ka];      a0.y = S[m][ka + 1];
    v2f a1; a1.x = S[16 + m][ka]; a1.y = S[16 + m][ka + 1];
#pragma unroll
    for (int nt = 0; nt < 8; ++nt) {
      const int n = nt * 16 + m;
      v2f b; b.x = Wp[(size_t)ka * HH + n]; b.y = Wp[(size_t)(ka + 1) * HH + n];
      acc0[nt] = wmma_f32(a0, b, acc0[nt]);
      acc1[nt] = wmma_f32(a1, b, acc1[nt]);
    }
  }

#pragma unroll
  for (int nt = 0; nt < 8; ++nt) {
    const int n = nt * 16 + m;
    const float bias = bp[n];
#pragma unroll
    for (int r = 0; r < 8; ++r) {
      const int rl0 = r + 8 * hf;        // local row, first half-tile
      const int rl1 = 16 + r + 8 * hf;   // local row, second half-tile
      out[(size_t)(n0 + rl0) * HH + n] = S[rl0][n] + acc0[nt][r] + bias;  // residual from LDS
      out[(size_t)(n0 + rl1) * HH + n] = S[rl1][n] + acc1[nt][r] + bias;
    }
  }
}

// ---------------- position update ----------------
__global__ void gate_kernel(const float* __restrict__ mes, const float* __restrict__ pw,
                            const float* __restrict__ pb, float* __restrict__ gate, int N) {
  const int node = blockIdx.x * (blockDim.x >> 5) + (threadIdx.x >> 5);
  const int lane = threadIdx.x & 31;
  if (node >= N) return;
  float s = 0.f;
  for (int c = lane; c < HH; c += 32) s += mes[(size_t)node * HH + c] * pw[c];
  s += __shfl_xor(s, 16); s += __shfl_xor(s, 8); s += __shfl_xor(s, 4);
  s += __shfl_xor(s, 2);  s += __shfl_xor(s, 1);
  if (lane == 0) gate[node] = s + pb[0];  // linear, no activation
}

__global__ void pos_edge_kernel(const float* __restrict__ pos, const int* __restrict__ adj,
                                const float* __restrict__ gate, float* __restrict__ acc, int E) {
  const int e = blockIdx.x * blockDim.x + threadIdx.x;
  if (e >= E) return;
  const int s = adj[e], r = adj[E + e];
  const float g = gate[r];
#pragma unroll
  for (int d = 0; d < 3; ++d)
    atomicAdd(&acc[(size_t)s * 3 + d], (pos[(size_t)s * 3 + d] - pos[(size_t)r * 3 + d]) * g);
}

__global__ void pos_final_kernel(const float* __restrict__ pos, const float* __restrict__ acc,
                                 float* __restrict__ out, int n3) {
  const int i = blockIdx.x * blockDim.x + threadIdx.x;
  if (i < n3) out[i] = pos[i] + POSC * acc[i];
}

// ---------------- launcher ----------------
extern "C" void kernel_launch(void* const* d_in, const int* in_sizes, int n_in,
                              void* d_out, int out_size, void* d_ws, size_t ws_size,
                              hipStream_t stream) {
  const float* x0  = (const float*)d_in[0];
  const float* x1  = (const float*)d_in[1];
  const float* pos = (const float*)d_in[2];

  const int*   adjA[3]; const float* invA[3]; const float* bngA[3]; const float* bnbA[3];
  const float* WA[3];   const float* bWA[3];  const float* iWA[3];  const float* ibA[3];
  for (int a = 0; a < 3; ++a) {
    const int base = 3 + a * 8;
    adjA[a] = (const int*)d_in[base + 0];
    invA[a] = (const float*)d_in[base + 1];
    bngA[a] = (const float*)d_in[base + 2];
    bnbA[a] = (const float*)d_in[base + 3];
    WA[a]   = (const float*)d_in[base + 4];
    bWA[a]  = (const float*)d_in[base + 5];
    iWA[a]  = (const float*)d_in[base + 6];
    ibA[a]  = (const float*)d_in[base + 7];
  }
  const float* u0g = (const float*)d_in[27];
  const float* u0b = (const float*)d_in[28];
  const float* u0W = (const float*)d_in[29];
  const float* u0bb= (const float*)d_in[30];
  const float* u1g = (const float*)d_in[31];
  const float* u1b = (const float*)d_in[32];
  const float* u1W = (const float*)d_in[33];
  const float* u1bb= (const float*)d_in[34];
  const float* posW= (const float*)d_in[35];
  const float* posb= (const float*)d_in[36];

  float* out_x0  = (float*)d_out;
  float* out_x1  = out_x0 + (size_t)NN * HH;
  float* out_pos = out_x1 + (size_t)NN * HH;

  // workspace layout (floats)
  float* ws = (float*)d_ws;
  size_t o = 0;
  float* mes[3]; for (int a = 0; a < 3; ++a) { mes[a] = ws + o; o += (size_t)NN * HH; }
  float* posacc = ws + o; o += (size_t)NN * 3;
  float* gate   = ws + o; o += NN;
  float* sums   = ws + o; o += 384;
  float* sqs    = ws + o; o += 384;
  float* Wp     = ws + o; o += (size_t)KPAD * HH;   // reused per adjacency
  float* bp     = ws + o; o += HH;
  float* uWp    = ws + o; o += (size_t)384 * HH;    // reused for u0/u1
  float* ubp    = ws + o; o += HH;

  // zero mes buffers + posacc + gate (contiguous)
  zero_kernel<<<2048, 256, 0, stream>>>(mes[0], (size_t)3 * NN * HH + (size_t)NN * 4);

  const float* sendp[3] = { x0, x0, x1 };
  const float* recp [3] = { x0, x1, x1 };

  for (int a = 0; a < 3; ++a) {
    zero_kernel<<<4, 256, 0, stream>>>(sums, 768);
    edge_stats_kernel<<<512, 288, 0, stream>>>(sendp[a], recp[a], adjA[a], invA[a],
                                               sums, sqs, EE);
    fold_kernel<<<1, 128, 0, stream>>>(WA[a], bWA[a], bngA[a], bnbA[a], sums, sqs,
                                       1.0f / (float)EE, Wp, bp, DINK, KPAD);
    msg_kernel<<<EE / 32, 32, 0, stream>>>(sendp[a], recp[a], adjA[a], invA[a],
                                           Wp, bp, iWA[a], ibA[a], mes[a], EE);
  }

  // update 0: concat(x0, mes00) -> x0n
  zero_kernel<<<4, 256, 0, stream>>>(sums, 768);
  node_stats_kernel<<<512, 256, 0, stream>>>(x0, mes[0], mes[0], sums, sqs, NN);
  fold_kernel<<<1, 128, 0, stream>>>(u0W, u0bb, u0g, u0b, sums, sqs,
                                     1.0f / (float)NN, uWp, ubp, 256, 256);
  update_kernel<<<NN / 32, 32, 0, stream>>>(x0, mes[0], mes[0], uWp, ubp, out_x0, 256);

  // update 1: concat(x1, mes01, mes11) -> x1n
  zero_kernel<<<4, 256, 0, stream>>>(sums, 768);
  node_stats_kernel<<<512, 384, 0, stream>>>(x1, mes[1], mes[2], sums, sqs, NN);
  fold_kernel<<<1, 128, 0, stream>>>(u1W, u1bb, u1g, u1b, sums, sqs,
                                     1.0f / (float)NN, uWp, ubp, 384, 384);
  update_kernel<<<NN / 32, 32, 0, stream>>>(x1, mes[1], mes[2], uWp, ubp, out_x1, 384);

  // position update via adj00
  gate_kernel<<<NN / 8, 256, 0, stream>>>(mes[0], posW, posb, gate, NN);
  pos_edge_kernel<<<(EE + 255) / 256, 256, 0, stream>>>(pos, adjA[0], gate, posacc, EE);
  pos_final_kernel<<<(3 * NN + 255) / 256, 256, 0, stream>>>(pos, posacc, out_pos, 3 * NN);
}